// EnvelopeLinearCQN_68135361184374
// MI455X (gfx1250) — compile-verified
//
#include <hip/hip_runtime.h>
#include <hip/hip_bf16.h>
#include <math.h>

// ---------------------------------------------------------------------------
// EnvelopeLinearCQN forward on MI455X (gfx1250, wave32, WMMA).
// Memory-bound on fp32 weight streaming (~350MB @ 23.3TB/s => ~15us floor).
// fp32 V_WMMA_F32_16X16X4_F32 for all GEMMs, gate GEMM pair fused, and
// triple-buffered GLOBAL_LOAD_ASYNC_TO_LDS_B128 pipelines: two tiles in
// flight per wave, partial S_WAIT_ASYNCCNT (in-order completion) so only the
// oldest tile is waited on. Grid is ordered M-fastest so the 8 batch-blocks
// sharing a weight N-strip are co-resident and L2 (192MB) deduplicates the
// weight stream to ~1x HBM traffic.
// ---------------------------------------------------------------------------

#define S_    126
#define R_    2
#define D_    128
#define H1_   5120
#define H2_   7680
#define B_    1024
#define A_    16
#define WNUM_ 4
#define SNUM_ (B_ / WNUM_)   // 256

typedef __attribute__((ext_vector_type(2))) float v2f;
typedef __attribute__((ext_vector_type(4))) float v4f;
typedef __attribute__((ext_vector_type(8))) float v8f;

__device__ __forceinline__ v8f wmma_f32(v2f a, v2f b, v8f c) {
  // D = A(16x4 f32) * B(4x16 f32) + C(16x16 f32)
  return __builtin_amdgcn_wmma_f32_16x16x4_f32(
      false, a, false, b, (short)0, c, false, false);
}

// Low 32 bits of a generic pointer into LDS == LDS byte offset.
__device__ __forceinline__ unsigned lds_off(const void* p) {
  return (unsigned)(unsigned long long)p;
}

// Async memory->LDS copy, 16B per lane. Tracked with ASYNCcnt (in-order).
__device__ __forceinline__ void async_g2l_b128(unsigned lds, const void* g) {
  asm volatile("global_load_async_to_lds_b128 %0, %1, off"
               :
               : "v"(lds), "v"(g)
               : "memory");
}

// Wait until at most `n` async ops remain outstanding (oldest completed).
#define WAIT_ASYNC(n) asm volatile("s_wait_asynccnt " #n ::: "memory")

// ------------------------- concat(state, preference) -----------------------
__global__ __launch_bounds__(256) void build_x_kernel(
    const float* __restrict__ state, const float* __restrict__ pref,
    float* __restrict__ X) {
  int idx = blockIdx.x * 256 + threadIdx.x;   // 0 .. B_*D_-1
  int row = idx >> 7;
  int col = idx & (D_ - 1);
  float v = (col < S_) ? state[row * S_ + col] : pref[row * R_ + (col - S_)];
  X[idx] = v;
}

// --------------------- GEMM + ReLU : C = relu(A@W + b) ---------------------
// Block tile 128x64, K-chunk 16, 8 waves each owning a 32x32 subtile.
// Triple-buffered async pipeline; 3 async ops per tile per wave.
__global__ __launch_bounds__(256) void gemm_relu_kernel(
    const float* __restrict__ Ag, const float* __restrict__ Wg,
    const float* __restrict__ bias, float* __restrict__ Cg,
    int N, int K) {
  __shared__ float As[3][128][20];   // row stride 80B (16B multiple)
  __shared__ float Bs[3][16][64];
  const unsigned ASTR = 128 * 20 * 4;
  const unsigned WSTR = 16 * 64 * 4;

  const int tid  = threadIdx.x;
  const int lane = tid & 31;
  const int wid  = tid >> 5;
  const int half = lane >> 4;
  const int l16  = lane & 15;
  const int m0 = blockIdx.x * 128;   // M fastest: weight-strip sharers adjacent
  const int n0 = blockIdx.y * 64;
  const int wm = (wid & 3) * 32;
  const int wn = (wid >> 2) * 32;

  const int ra = tid >> 2, ca = (tid & 3) << 2;     // A rows ra, ra+64
  const int rw = tid >> 4, cw = (tid & 15) << 2;    // W row rw

  const float* gA0 = Ag + (size_t)(m0 + ra) * K + ca;
  const float* gA1 = Ag + (size_t)(m0 + ra + 64) * K + ca;
  const float* gW  = Wg + (size_t)rw * N + n0 + cw;

  const unsigned lA0 = lds_off(&As[0][ra][ca]);
  const unsigned lA1 = lds_off(&As[0][ra + 64][ca]);
  const unsigned lW  = lds_off(&Bs[0][rw][cw]);

  v8f acc[2][2] = {};

  auto issue = [&](unsigned buf) {
    async_g2l_b128(lA0 + buf * ASTR, gA0);
    async_g2l_b128(lA1 + buf * ASTR, gA1);
    async_g2l_b128(lW  + buf * WSTR, gW);
    gA0 += 16; gA1 += 16; gW += (size_t)16 * N;
  };
  auto compute = [&](int cur) {
#pragma unroll
    for (int kk = 0; kk < 16; kk += 4) {
      v2f a0, a1, b0, b1;
      a0.x = As[cur][wm + l16][kk + 2 * half];
      a0.y = As[cur][wm + l16][kk + 2 * half + 1];
      a1.x = As[cur][wm + 16 + l16][kk + 2 * half];
      a1.y = As[cur][wm + 16 + l16][kk + 2 * half + 1];
      b0.x = Bs[cur][kk + 2 * half][wn + l16];
      b0.y = Bs[cur][kk + 2 * half + 1][wn + l16];
      b1.x = Bs[cur][kk + 2 * half][wn + 16 + l16];
      b1.y = Bs[cur][kk + 2 * half + 1][wn + 16 + l16];
      acc[0][0] = wmma_f32(a0, b0, acc[0][0]);
      acc[0][1] = wmma_f32(a0, b1, acc[0][1]);
      acc[1][0] = wmma_f32(a1, b0, acc[1][0]);
      acc[1][1] = wmma_f32(a1, b1, acc[1][1]);
    }
  };

  const int nchunks = K / 16;        // >= 8 for all launches
  issue(0);
  issue(1);
  int cur = 0, nxt = 2;
  for (int it = 0; it < nchunks - 1; ++it) {
    WAIT_ASYNC(3);                   // oldest tile (3 ops) landed
    __syncthreads();
    if (it + 2 < nchunks) issue(nxt);
    compute(cur);
    cur = (cur == 2) ? 0 : cur + 1;
    nxt = (nxt == 2) ? 0 : nxt + 1;
  }
  WAIT_ASYNC(0);
  __syncthreads();
  compute(cur);

#pragma unroll
  for (int sm = 0; sm < 2; ++sm)
#pragma unroll
    for (int sn = 0; sn < 2; ++sn) {
      int col = n0 + wn + sn * 16 + l16;
      float bv = bias[col];
#pragma unroll
      for (int r = 0; r < 8; ++r) {
        int row = m0 + wm + sm * 16 + half * 8 + r;
        float v = acc[sm][sn][r] + bv;
        Cg[(size_t)row * N + col] = fmaxf(v, 0.0f);
      }
    }
}

// ------- Fused dual GEMM gate: out = relu(A@W2+b2) * sigmoid(500*(A@Wg+bg))
__global__ __launch_bounds__(256) void gemm_gate_kernel(
    const float* __restrict__ Ag,
    const float* __restrict__ Wgate, const float* __restrict__ bgate,
    const float* __restrict__ W2, const float* __restrict__ b2,
    float* __restrict__ Cg, int N, int K) {
  __shared__ float As[3][128][20];
  __shared__ float Bg[3][16][64];
  __shared__ float Bh[3][16][64];
  const unsigned ASTR = 128 * 20 * 4;
  const unsigned WSTR = 16 * 64 * 4;

  const int tid  = threadIdx.x;
  const int lane = tid & 31;
  const int wid  = tid >> 5;
  const int half = lane >> 4;
  const int l16  = lane & 15;
  const int m0 = blockIdx.x * 128;
  const int n0 = blockIdx.y * 64;
  const int wm = (wid & 3) * 32;
  const int wn = (wid >> 2) * 32;

  const int ra = tid >> 2, ca = (tid & 3) << 2;
  const int rw = tid >> 4, cw = (tid & 15) << 2;

  const float* gA0 = Ag    + (size_t)(m0 + ra) * K + ca;
  const float* gA1 = Ag    + (size_t)(m0 + ra + 64) * K + ca;
  const float* gWg = Wgate + (size_t)rw * N + n0 + cw;
  const float* gWh = W2    + (size_t)rw * N + n0 + cw;

  const unsigned lA0 = lds_off(&As[0][ra][ca]);
  const unsigned lA1 = lds_off(&As[0][ra + 64][ca]);
  const unsigned lBg = lds_off(&Bg[0][rw][cw]);
  const unsigned lBh = lds_off(&Bh[0][rw][cw]);

  v8f accg[2][2] = {};
  v8f acch[2][2] = {};

  auto issue = [&](unsigned buf) {
    async_g2l_b128(lA0 + buf * ASTR, gA0);
    async_g2l_b128(lA1 + buf * ASTR, gA1);
    async_g2l_b128(lBg + buf * WSTR, gWg);
    async_g2l_b128(lBh + buf * WSTR, gWh);
    gA0 += 16; gA1 += 16; gWg += (size_t)16 * N; gWh += (size_t)16 * N;
  };
  auto compute = [&](int cur) {
#pragma unroll
    for (int kk = 0; kk < 16; kk += 4) {
      v2f a0, a1, g0, g1, h0, h1;
      a0.x = As[cur][wm + l16][kk + 2 * half];
      a0.y = As[cur][wm + l16][kk + 2 * half + 1];
      a1.x = As[cur][wm + 16 + l16][kk + 2 * half];
      a1.y = As[cur][wm + 16 + l16][kk + 2 * half + 1];
      g0.x = Bg[cur][kk + 2 * half][wn + l16];
      g0.y = Bg[cur][kk + 2 * half + 1][wn + l16];
      g1.x = Bg[cur][kk + 2 * half][wn + 16 + l16];
      g1.y = Bg[cur][kk + 2 * half + 1][wn + 16 + l16];
      h0.x = Bh[cur][kk + 2 * half][wn + l16];
      h0.y = Bh[cur][kk + 2 * half + 1][wn + l16];
      h1.x = Bh[cur][kk + 2 * half][wn + 16 + l16];
      h1.y = Bh[cur][kk + 2 * half + 1][wn + 16 + l16];
      accg[0][0] = wmma_f32(a0, g0, accg[0][0]);
      accg[0][1] = wmma_f32(a0, g1, accg[0][1]);
      accg[1][0] = wmma_f32(a1, g0, accg[1][0]);
      accg[1][1] = wmma_f32(a1, g1, accg[1][1]);
      acch[0][0] = wmma_f32(a0, h0, acch[0][0]);
      acch[0][1] = wmma_f32(a0, h1, acch[0][1]);
      acch[1][0] = wmma_f32(a1, h0, acch[1][0]);
      acch[1][1] = wmma_f32(a1, h1, acch[1][1]);
    }
  };

  const int nchunks = K / 16;        // 320
  issue(0);
  issue(1);
  int cur = 0, nxt = 2;
  for (int it = 0; it < nchunks - 1; ++it) {
    WAIT_ASYNC(4);                   // oldest tile (4 ops) landed
    __syncthreads();
    if (it + 2 < nchunks) issue(nxt);
    compute(cur);
    cur = (cur == 2) ? 0 : cur + 1;
    nxt = (nxt == 2) ? 0 : nxt + 1;
  }
  WAIT_ASYNC(0);
  __syncthreads();
  compute(cur);

#pragma unroll
  for (int sm = 0; sm < 2; ++sm)
#pragma unroll
    for (int sn = 0; sn < 2; ++sn) {
      int col = n0 + wn + sn * 16 + l16;
      float bgv = bgate[col];
      float b2v = b2[col];
#pragma unroll
      for (int r = 0; r < 8; ++r) {
        int row = m0 + wm + sm * 16 + half * 8 + r;
        float z = (accg[sm][sn][r] + bgv) * 500.0f;
        float gate = 1.0f / (1.0f + __expf(-z));
        float h = fmaxf(acch[sm][sn][r] + b2v, 0.0f);
        Cg[(size_t)row * N + col] = h * gate;
      }
    }
}

// ---------------------- GEMM3 : q = A@W3 + b3 (N = 32) ---------------------
// Block tile 128x32, K-chunk 32, 8 waves each owning a 16x32 row band.
__global__ __launch_bounds__(256) void gemm_q_kernel(
    const float* __restrict__ Ag, const float* __restrict__ W3,
    const float* __restrict__ bias, float* __restrict__ Cg, int K) {
  const int N = A_ * R_;             // 32
  __shared__ float As[3][128][36];   // row stride 144B (16B multiple)
  __shared__ float Bs[3][32][32];
  const unsigned ASTR = 128 * 36 * 4;
  const unsigned WSTR = 32 * 32 * 4;

  const int tid  = threadIdx.x;
  const int lane = tid & 31;
  const int wid  = tid >> 5;
  const int half = lane >> 4;
  const int l16  = lane & 15;
  const int m0 = blockIdx.x * 128;
  const int wm = wid * 16;

  const int ra = tid >> 3, ca = (tid & 7) << 2;   // A rows ra + i*32
  const int rw = tid >> 3, cw = (tid & 7) << 2;

  const float* gA0 = Ag + (size_t)(m0 + ra) * K + ca;
  const float* gA1 = Ag + (size_t)(m0 + ra + 32) * K + ca;
  const float* gA2 = Ag + (size_t)(m0 + ra + 64) * K + ca;
  const float* gA3 = Ag + (size_t)(m0 + ra + 96) * K + ca;
  const float* gW  = W3 + (size_t)rw * N + cw;

  const unsigned lA0 = lds_off(&As[0][ra][ca]);
  const unsigned lA1 = lds_off(&As[0][ra + 32][ca]);
  const unsigned lA2 = lds_off(&As[0][ra + 64][ca]);
  const unsigned lA3 = lds_off(&As[0][ra + 96][ca]);
  const unsigned lW  = lds_off(&Bs[0][rw][cw]);

  v8f acc0 = {}, acc1 = {};

  auto issue = [&](unsigned buf) {
    async_g2l_b128(lA0 + buf * ASTR, gA0);
    async_g2l_b128(lA1 + buf * ASTR, gA1);
    async_g2l_b128(lA2 + buf * ASTR, gA2);
    async_g2l_b128(lA3 + buf * ASTR, gA3);
    async_g2l_b128(lW  + buf * WSTR, gW);
    gA0 += 32; gA1 += 32; gA2 += 32; gA3 += 32;
    gW += (size_t)32 * N;
  };
  auto compute = [&](int cur) {
#pragma unroll
    for (int kk = 0; kk < 32; kk += 4) {
      v2f a, b0, b1;
      a.x = As[cur][wm + l16][kk + 2 * half];
      a.y = As[cur][wm + l16][kk + 2 * half + 1];
      b0.x = Bs[cur][kk + 2 * half][l16];
      b0.y = Bs[cur][kk + 2 * half + 1][l16];
      b1.x = Bs[cur][kk + 2 * half][16 + l16];
      b1.y = Bs[cur][kk + 2 * half + 1][16 + l16];
      acc0 = wmma_f32(a, b0, acc0);
      acc1 = wmma_f32(a, b1, acc1);
    }
  };

  const int nchunks = K / 32;        // 240
  issue(0);
  issue(1);
  int cur = 0, nxt = 2;
  for (int it = 0; it < nchunks - 1; ++it) {
    WAIT_ASYNC(5);                   // oldest tile (5 ops) landed
    __syncthreads();
    if (it + 2 < nchunks) issue(nxt);
    compute(cur);
    cur = (cur == 2) ? 0 : cur + 1;
    nxt = (nxt == 2) ? 0 : nxt + 1;
  }
  WAIT_ASYNC(0);
  __syncthreads();
  compute(cur);

#pragma unroll
  for (int r = 0; r < 8; ++r) {
    int row = m0 + wm + half * 8 + r;
    int c0 = l16, c1 = 16 + l16;
    Cg[(size_t)row * N + c0] = acc0[r] + bias[c0];
    Cg[(size_t)row * N + c1] = acc1[r] + bias[c1];
  }
}

// ------------- Envelope top-1 scalarized Q selection (hq output) -----------
__global__ __launch_bounds__(256) void envelope_kernel(
    const float* __restrict__ q, const float* __restrict__ pref,
    float* __restrict__ hq) {
  const int tid  = threadIdx.x;
  const int lane = tid & 31;
  const int wid  = tid >> 5;
  const int p = blockIdx.x * 8 + wid;          // one wave per output row
  const int grp = p & (SNUM_ - 1);

  float p0 = pref[p * R_ + 0];
  float p1 = pref[p * R_ + 1];

  float best = -3.402823466e+38f;
  int   bidx = 0x7fffffff;
  float br0 = 0.0f, br1 = 0.0f;

#pragma unroll
  for (int i = 0; i < 2; ++i) {
    int k = lane * 2 + i;                      // candidate 0..63
    int j = k >> 4;                            // preference-copy index
    int act = k & (A_ - 1);
    int row = grp + j * SNUM_;
    float r0 = q[((size_t)row * A_ + act) * R_ + 0];
    float r1 = q[((size_t)row * A_ + act) * R_ + 1];
    float sc = r0 * p0 + r1 * p1;
    if (sc > best || (sc == best && k < bidx)) {
      best = sc; bidx = k; br0 = r0; br1 = r1;
    }
  }
  for (int off = 16; off > 0; off >>= 1) {
    float ob = __shfl_xor(best, off, 32);
    int   oi = __shfl_xor(bidx, off, 32);
    float o0 = __shfl_xor(br0, off, 32);
    float o1 = __shfl_xor(br1, off, 32);
    if (ob > best || (ob == best && oi < bidx)) {
      best = ob; bidx = oi; br0 = o0; br1 = o1;
    }
  }
  if (lane == 0) {
    hq[p * R_ + 0] = br0;
    hq[p * R_ + 1] = br1;
  }
}

// ---------------------------------------------------------------------------
extern "C" void kernel_launch(void* const* d_in, const int* in_sizes, int n_in,
                              void* d_out, int out_size, void* d_ws,
                              size_t ws_size, hipStream_t stream) {
  const float* state = (const float*)d_in[0];
  const float* pref  = (const float*)d_in[1];
  const float* W1 = (const float*)d_in[2];
  const float* b1 = (const float*)d_in[3];
  const float* Wg = (const float*)d_in[4];
  const float* bg = (const float*)d_in[5];
  const float* W2 = (const float*)d_in[6];
  const float* b2 = (const float*)d_in[7];
  const float* W3 = (const float*)d_in[8];
  const float* b3 = (const float*)d_in[9];
  float* out = (float*)d_out;

  char* ws = (char*)d_ws;
  float* X   = (float*)(ws);                                   // 512 KiB
  float* H1  = (float*)(ws + (size_t)(1 << 20));               // 20 MiB
  float* ACT = (float*)(ws + (size_t)(21) * (1 << 20));        // 30 MiB

  float* q  = out + (size_t)B_ * R_;   // d_out = [hq (2048) | q (32768)]
  float* hq = out;

  build_x_kernel<<<(B_ * D_) / 256, 256, 0, stream>>>(state, pref, X);

  // M (batch) is the fast grid dimension: the 8 blocks sharing each weight
  // N-strip launch back-to-back so L2 serves 7 of 8 weight reads.
  gemm_relu_kernel<<<dim3(B_ / 128, H1_ / 64), 256, 0, stream>>>(
      X, W1, b1, H1, H1_, D_);

  gemm_gate_kernel<<<dim3(B_ / 128, H2_ / 64), 256, 0, stream>>>(
      H1, Wg, bg, W2, b2, ACT, H2_, H1_);

  gemm_q_kernel<<<dim3(B_ / 128), 256, 0, stream>>>(ACT, W3, b3, q, H2_);

  envelope_kernel<<<dim3(B_ / 8), 256, 0, stream>>>(q, pref, hq);
}